// Slice_40853728920245
// MI455X (gfx1250) — compile-verified
//
#include <hip/hip_runtime.h>
#include <stdint.h>

// Problem constants (shapes fixed by the reference)
#define GD   8      // grid depth  (z <- width coord)
#define GH   16     // grid height (y <- height coord)
#define GW   16     // grid width  (x <- guide value)
#define NC   12     // channels
#define IMH  1080
#define IMW  1920
#define BLOCK 256   // 8 waves (wave32)
#define VEC   4     // pixels per lane per chunk (b128 IO)
#define CHUNK (BLOCK * VEC)      // 1024 consecutive pixels per chunk
#define CPB   2                  // chunks per block -> 2048 px/block
#define GRID_ELEMS (NC * GD * GH * GW)       // 24576 floats = 96 KB (fits 320KB WGP LDS)
#define CH_STRIDE  (GD * GH * GW)            // 2048 floats per channel plane

typedef uint32_t u32x4 __attribute__((ext_vector_type(4)));
typedef uint32_t u32x8 __attribute__((ext_vector_type(8)));
typedef float    f32x4 __attribute__((ext_vector_type(4)));

__device__ __forceinline__ int clampi(int v, int lo, int hi) {
    return v < lo ? lo : (v > hi ? hi : v);
}

// ---------------------------------------------------------------------------
// Tensor Data Mover: 1-D tile copy, global -> LDS, data_size = 4 bytes.
// D# per CDNA5 ISA ch.8 (verified lowering: tensor_load_to_lds s[g0], s[g1]).
// ---------------------------------------------------------------------------
__device__ __forceinline__ void tdm_load_1d_to_lds(const void* gptr,
                                                   uint32_t lds_byte_addr,
                                                   uint32_t nelems) {
    uint64_t ga = (uint64_t)(uintptr_t)gptr;
    u32x4 g0;
    g0.s0 = 1u;                                                  // count=1, user desc
    g0.s1 = lds_byte_addr;                                       // lds_addr
    g0.s2 = (uint32_t)ga;                                        // global_addr[31:0]
    g0.s3 = ((uint32_t)(ga >> 32) & 0x01FFFFFFu) | (2u << 30);   // ga[56:32] | type=2

    u32x8 g1;
    g1.s0 = (2u << 16);                         // workgroup_mask=0, data_size=2 (4B)
    g1.s1 = (nelems & 0xFFFFu) << 16;           // tensor_dim0[15:0]  (bits 63:48)
    g1.s2 = (nelems >> 16) | (1u << 16);        // tensor_dim0[31:16] | tensor_dim1=1
    g1.s3 = (nelems & 0xFFFFu) << 16;           // tile_dim0 (bits 127:112)
    g1.s4 = 0u;                                 // tile_dim1=0, tile_dim2=0 (1-D)
    g1.s5 = nelems;                             // tensor_dim0_stride[31:0]
    g1.s6 = 0u;
    g1.s7 = 0u;

    asm volatile("tensor_load_to_lds %0, %1" :: "s"(g0), "s"(g1) : "memory");
}

// Per-pixel trilinear setup: 4 LDS pair-bases + 8 masked weights.
// x handled as adjacent pair (xb, xb+1) with branchless re-weighting that is
// bit-equivalent to grid_sample's clamp+zero-mask semantics.
__device__ __forceinline__ void setup_pixel(float gx, int h, int w,
                                            int* __restrict__ b,
                                            float* __restrict__ Wt) {
    const float sy = 2.0f / (float)(IMH - 1);
    const float sz = 2.0f / (float)(IMW - 1);
    float yy = (float)h * sy - 1.0f;
    float zz = (float)w * sz - 1.0f;
    float ix = ((gx + 1.0f) * 16.0f - 1.0f) * 0.5f;
    float iy = ((yy + 1.0f) * 16.0f - 1.0f) * 0.5f;
    float iz = ((zz + 1.0f) * 8.0f  - 1.0f) * 0.5f;

    float fxf = floorf(ix), fyf = floorf(iy), fzf = floorf(iz);
    float fx = ix - fxf, fy = iy - fyf, fz = iz - fzf;
    int x0 = (int)fxf, y0 = (int)fyf, z0 = (int)fzf;
    int x1 = x0 + 1, y1 = y0 + 1, z1 = z0 + 1;

    float wy0 = (1.0f - fy) * ((y0 >= 0 && y0 < GH) ? 1.0f : 0.0f);
    float wy1 = fy          * ((y1 >= 0 && y1 < GH) ? 1.0f : 0.0f);
    float wz0 = (1.0f - fz) * ((z0 >= 0 && z0 < GD) ? 1.0f : 0.0f);
    float wz1 = fz          * ((z1 >= 0 && z1 < GD) ? 1.0f : 0.0f);
    int cy0 = clampi(y0, 0, GH - 1), cy1 = clampi(y1, 0, GH - 1);
    int cz0 = clampi(z0, 0, GD - 1), cz1 = clampi(z1, 0, GD - 1);

    int xb = clampi(x0, 0, GW - 2);
    float w_lo = 1.0f - fx, w_hi = fx;
    float a_lo = (xb     == x0) ? w_lo : ((xb     == x1) ? w_hi : 0.0f);
    float a_hi = (xb + 1 == x0) ? w_lo : ((xb + 1 == x1) ? w_hi : 0.0f);

    b[0] = (cz0 * GH + cy0) * GW + xb;
    b[1] = (cz0 * GH + cy1) * GW + xb;
    b[2] = (cz1 * GH + cy0) * GW + xb;
    b[3] = (cz1 * GH + cy1) * GW + xb;

    float w00 = wz0 * wy0, w01 = wz0 * wy1;
    float w10 = wz1 * wy0, w11 = wz1 * wy1;
    Wt[0] = w00 * a_lo;  Wt[1] = w00 * a_hi;
    Wt[2] = w01 * a_lo;  Wt[3] = w01 * a_hi;
    Wt[4] = w10 * a_lo;  Wt[5] = w10 * a_hi;
    Wt[6] = w11 * a_lo;  Wt[7] = w11 * a_hi;
}

__device__ __forceinline__ float gather8(const float* __restrict__ S,
                                         const int* __restrict__ b,
                                         const float* __restrict__ Wt) {
    float a;
    a  = Wt[0] * S[b[0]] + Wt[1] * S[b[0] + 1];   // pairs fuse into ds read2
    a += Wt[2] * S[b[1]] + Wt[3] * S[b[1] + 1];
    a += Wt[4] * S[b[2]] + Wt[5] * S[b[2] + 1];
    a += Wt[6] * S[b[3]] + Wt[7] * S[b[3] + 1];
    return a;
}

__global__ __launch_bounds__(BLOCK) void bilateral_slice_kernel(
    const float* __restrict__ grid,    // [12, 8, 16, 16]
    const float* __restrict__ guide,   // [1080, 1920]
    float* __restrict__ out,           // [12, 1080, 1920]
    int P)                             // 1080*1920 (multiple of 4)
{
    __shared__ float sgrid[GRID_ELEMS];

    // Wave 0 DMAs the whole 96KB grid into LDS via the TDM (TENSORcnt path);
    // other waves go straight to the barrier.
    if (threadIdx.x < 32u) {
        tdm_load_1d_to_lds(grid, (uint32_t)(uintptr_t)&sgrid[0], GRID_ELEMS);
        __builtin_amdgcn_s_wait_tensorcnt(0);
    }
    __syncthreads();

    #pragma unroll 1
    for (int i = 0; i < CPB; ++i) {
        int p4 = (blockIdx.x * CPB + i) * CHUNK + (int)threadIdx.x * VEC;
        if (p4 >= P) break;            // P % VEC == 0: whole-vector tail guard

        // 4 guide values per lane, non-temporal b128 load
        f32x4 g = __builtin_nontemporal_load((const f32x4*)(guide + p4));

        int   bs[VEC][4];
        float wt[VEC][8];
        int h = p4 / IMW;
        int w = p4 - h * IMW;
        #pragma unroll
        for (int j = 0; j < VEC; ++j) {
            setup_pixel(g[j], h, w, bs[j], wt[j]);
            ++w;
            if (w == IMW) { w = 0; ++h; }   // row crossing inside the quad
        }

        #pragma unroll
        for (int c = 0; c < NC; ++c) {
            const float* S = sgrid + c * CH_STRIDE;
            f32x4 acc;
            acc.x = gather8(S, bs[0], wt[0]);
            acc.y = gather8(S, bs[1], wt[1]);
            acc.z = gather8(S, bs[2], wt[2]);
            acc.w = gather8(S, bs[3], wt[3]);
            // coalesced, aligned, streaming b128 store per channel plane
            __builtin_nontemporal_store(acc, (f32x4*)(out + c * P + p4));
        }
    }
}

extern "C" void kernel_launch(void* const* d_in, const int* in_sizes, int n_in,
                              void* d_out, int out_size, void* d_ws, size_t ws_size,
                              hipStream_t stream) {
    const float* grid  = (const float*)d_in[0];   // 24576 f32
    const float* guide = (const float*)d_in[1];   // 2073600 f32
    float* out = (float*)d_out;                   // 12 * 2073600 f32
    int P = in_sizes[1];
    int blocks = (P + CHUNK * CPB - 1) / (CHUNK * CPB);   // 1013
    bilateral_slice_kernel<<<blocks, BLOCK, 0, stream>>>(grid, guide, out, P);
}